// HeatmapOffsetmapLoss_88802743812713
// MI455X (gfx1250) — compile-verified
//
#include <hip/hip_runtime.h>

// ---------------------------------------------------------------------------
// HeatmapOffsetmapLoss for MI455X (gfx1250, wave32)
//
// Strategy: the analytic heat/offset maps collapse to closed forms in local
// coords, so this is a pure 239MB streaming reduction (HBM floor ~10.3us at
// 23.3 TB/s). Kernel 1 streams float4s (global_load_b128) with a one-stride-
// ahead global_prefetch_b8, computes per-element loss terms with fast
// v_exp_f32/v_log_f32 softplus, reduces per block (wave32 shfl_xor + LDS) and
// writes one partial per block to workspace. Kernel 2 (1 wave) reduces the
// 2048 partials deterministically, using V_WMMA_F32_16X16X4_F32 with an
// all-ones B matrix as the cross-lane summation engine (D[m,n] = p_m + p_m+16),
// then normalizes by (H*W * mask_sum).
// ---------------------------------------------------------------------------

typedef __attribute__((ext_vector_type(2))) float v2f;
typedef __attribute__((ext_vector_type(8))) float v8f;

#define HH 512
#define WW 512
#define NLM 19
#define NCH 57
#define NBATCH 4

constexpr int   NB    = 2048;                       // reduction blocks
constexpr int   NT    = 256;                        // threads / block (8 waves)
constexpr long  NPIX  = (long)HH * WW;              // 262144
constexpr long  NELEM = (long)NBATCH * NCH * NPIX;  // 59,768,832
constexpr long  NV4   = NELEM / 4;                  // float4 count

__global__ __launch_bounds__(NT)
void loss_partials_kernel(const float4* __restrict__ fm,
                          const int*    __restrict__ lm,
                          float*        __restrict__ partials) {
    __shared__ int slmx[NBATCH * NLM];
    __shared__ int slmy[NBATCH * NLM];
    const int tid = threadIdx.x;
    if (tid < NBATCH * NLM) {
        slmx[tid] = lm[2 * tid + 0];
        slmy[tid] = lm[2 * tid + 1];
    }
    __syncthreads();

    const long stride = (long)gridDim.x * blockDim.x;
    float acc = 0.f;

    for (long v = (long)blockIdx.x * blockDim.x + tid; v < NV4; v += stride) {
        // Speculative prefetch one grid-stride ahead (gfx1250 global_prefetch_b8;
        // OOB speculative prefetches are silently dropped).
        __builtin_prefetch(&fm[v + stride], 0, 1);

        const float4 z4 = fm[v];
        const long e = v << 2;                 // flat element index of .x
        const int  j0 = (int)(e & (WW - 1));   // 512 | 4 -> all 4 share row/chan
        const long t  = e >> 9;
        const int  i  = (int)(t & (HH - 1));
        const int  u  = (int)(t >> 9);         // b*57 + c, < 228
        const int  c  = u % NCH;
        const int  b  = u / NCH;
        const int  grp = c / NLM;              // 0: heat, 1: off_x, 2: off_y
        const int  n   = c - grp * NLM;

        const int xr = slmx[b * NLM + n];
        const int yr = slmy[b * NLM + n];
        const float msk = (xr > 0 && yr > 0) ? 1.f : 0.f;
        const int x = min(max(xr, 0), WW - 1);
        const int y = min(max(yr, 0), HH - 1);

        const float zz[4] = {z4.x, z4.y, z4.z, z4.w};
        float s = 0.f;

        if (grp == 0) {
            // heat[i,j] = exp(-0.5*d2/40^2) * [d2 <= 1600], d2=(i-y)^2+(j-x)^2
            const float dy  = (float)(i - y);
            const float dy2 = dy * dy;
            #pragma unroll
            for (int q = 0; q < 4; ++q) {
                const float dx = (float)(j0 + q - x);
                const float d2 = dy2 + dx * dx;
                float hm = 0.f;
                if (d2 <= 1600.f) hm = __expf(d2 * (-0.5f / 1600.f));
                const float z  = zz[q];
                // stable softplus via fast transcendentals
                const float sp = fmaxf(z, 0.f) + __logf(1.f + __expf(-fabsf(z)));
                s += 2.f * (sp - z * hm);
            }
        } else if (grp == 1) {
            // off_x target = (y - i)/40  (constant across the 4 columns)
            const float omx = (float)(y - i) * (1.f / 40.f);
            #pragma unroll
            for (int q = 0; q < 4; ++q) s += fabsf(zz[q] - omx);
        } else {
            // off_y target = (512 - x + j)/40
            #pragma unroll
            for (int q = 0; q < 4; ++q) {
                const float omy = (float)(WW - x + j0 + q) * (1.f / 40.f);
                s += fabsf(zz[q] - omy);
            }
        }
        acc += msk * s;
    }

    // wave32 butterfly reduction
    #pragma unroll
    for (int o = 16; o > 0; o >>= 1) acc += __shfl_xor(acc, o, 32);

    __shared__ float wsum[NT / 32];
    const int lane = tid & 31, wid = tid >> 5;
    if (lane == 0) wsum[wid] = acc;
    __syncthreads();
    if (tid == 0) {
        float sblk = 0.f;
        #pragma unroll
        for (int w = 0; w < NT / 32; ++w) sblk += wsum[w];
        partials[blockIdx.x] = sblk;
    }
}

// Single-wave finalize: deterministic serial lane sums, then a WMMA f32
// 16x16x4 against an all-ones B matrix performs the cross-lane reduction.
__global__ __launch_bounds__(32)
void loss_finalize_kernel(const float* __restrict__ partials,
                          const int*   __restrict__ lm,
                          float*       __restrict__ out) {
    const int lane = threadIdx.x;

    float p = 0.f;
    for (int k = lane; k < NB; k += 32) p += partials[k];   // fixed order

    // A (16x4 f32): lane L<16 holds A[L,0]=p_L (VGPR0), A[L,1]=0 (VGPR1);
    // lane L>=16 holds A[L-16,2]=p_L, A[L-16,3]=0.  B = ones(4x16).
    // => D[m,n] = p_m + p_{m+16} for every n.
    v2f a;  a.x = p;   a.y = 0.f;
    v2f bm; bm.x = 1.f; bm.y = 1.f;
    v8f c = {};
    c = __builtin_amdgcn_wmma_f32_16x16x4_f32(
            /*neg_a=*/false, a, /*neg_b=*/false, bm,
            /*c_mod=*/(short)0, c, /*reuse_a=*/false, /*reuse_b=*/false);

    // Lane l<16 rows M=0..7, lane l>=16 rows M=8..15: sum VGPRs then fold halves.
    float s = c[0] + c[1] + c[2] + c[3] + c[4] + c[5] + c[6] + c[7];
    s += __shfl_xor(s, 16, 32);      // total = sum of all 32 lane partials

    // mask_sum recomputed redundantly per lane (76 cached loads, trivial)
    float msum = 0.f;
    for (int k = 0; k < NBATCH * NLM; ++k) {
        const int xr = lm[2 * k + 0];
        const int yr = lm[2 * k + 1];
        msum += (xr > 0 && yr > 0) ? 1.f : 0.f;
    }

    if (lane == 0) out[0] = s / ((float)NPIX * msum);
}

extern "C" void kernel_launch(void* const* d_in, const int* in_sizes, int n_in,
                              void* d_out, int out_size, void* d_ws, size_t ws_size,
                              hipStream_t stream) {
    const float* fm  = (const float*)d_in[0];   // (4, 57, 512, 512) f32
    const int*   lm  = (const int*)d_in[1];     // (4, 19, 2) i32
    float*       out = (float*)d_out;           // scalar f32
    float*       partials = (float*)d_ws;       // NB floats (8 KB) of scratch

    loss_partials_kernel<<<NB, NT, 0, stream>>>((const float4*)fm, lm, partials);
    loss_finalize_kernel<<<1, 32, 0, stream>>>(partials, lm, out);
}